// SemanticDecouplingOriginal_83820581749245
// MI455X (gfx1250) — compile-verified
//
#include <hip/hip_runtime.h>
#include <hip/hip_bf16.h>
#include <math.h>

// Problem constants (from reference)
#define BB   8
#define CC   2048
#define HW   196      // 14*14
#define KK   80
#define WDIM 300
#define DD   1024
#define MM   1568     // BB*HW

typedef __attribute__((ext_vector_type(2))) float v2f;
typedef __attribute__((ext_vector_type(4))) float v4f;
typedef __attribute__((ext_vector_type(8))) float v8f;

#if __has_builtin(__builtin_amdgcn_tanhf)
#define FAST_TANH(x) __builtin_amdgcn_tanhf(x)
#else
#define FAST_TANH(x) tanhf(x)
#endif

// CDNA5 async global->LDS copy (ASYNCcnt-tracked, no VGPR round trip).
__device__ __forceinline__ void async_ld128(unsigned ldsOff, const void* gptr) {
    asm volatile("global_load_async_to_lds_b128 %0, %1, off"
                 :: "v"(ldsOff), "v"((unsigned long long)(size_t)gptr)
                 : "memory");
}
__device__ __forceinline__ void wait_async_le(void) {   // all async done
    asm volatile("s_wait_asynccnt 0x0" ::: "memory");
}
__device__ __forceinline__ void wait_async_le2(void) {  // keep newest 2 in flight
    asm volatile("s_wait_asynccnt 0x2" ::: "memory");
}

// ---------------------------------------------------------------------------
// 1) Transpose img (B, C, HW) NCHW -> xT (B, HW, C): coalesced downstream
//    reads along C; xT (12.8 MB) stays resident in the 192 MB L2.
// ---------------------------------------------------------------------------
__global__ void k_transpose(const float* __restrict__ img, float* __restrict__ xT) {
    __shared__ float tile[32][33];
    int b   = blockIdx.z;
    int hw0 = blockIdx.x * 32;
    int c0  = blockIdx.y * 32;
    int tx = threadIdx.x;   // 32
    int ty = threadIdx.y;   // 8
    for (int i = 0; i < 32; i += 8) {
        int c  = c0 + ty + i;
        int hw = hw0 + tx;
        float v = 0.f;
        if (hw < HW) v = img[(size_t)b * CC * HW + (size_t)c * HW + hw];
        tile[ty + i][tx] = v;
    }
    __syncthreads();
    for (int i = 0; i < 32; i += 8) {
        int hw = hw0 + ty + i;
        int c  = c0 + tx;
        if (hw < HW) xT[((size_t)b * HW + hw) * CC + c] = tile[tx][ty + i];
    }
}

// ---------------------------------------------------------------------------
// 2) Fold fc3+fc4: v[d] = sum_e W4[e]*W3[e][d]  (kills the 263-GFLOP GEMM).
// ---------------------------------------------------------------------------
__global__ void k_foldW34(const float* __restrict__ W3, const float* __restrict__ W4,
                          float* __restrict__ v) {
    int d = blockIdx.x * blockDim.x + threadIdx.x;   // 0..1023
    float acc = 0.f;
    for (int e = 0; e < DD; ++e)
        acc = fmaf(W4[e], W3[(size_t)e * DD + d], acc);
    v[d] = acc;
}

// c0 = sum_e b3[e]*W4[e] + b4[0]
__global__ void k_c0(const float* __restrict__ b3, const float* __restrict__ W4,
                     const float* __restrict__ b4, float* __restrict__ c0) {
    __shared__ float red[256];
    int t = threadIdx.x;
    float s = 0.f;
    for (int e = t; e < DD; e += 256) s = fmaf(b3[e], W4[e], s);
    red[t] = s;
    __syncthreads();
    for (int off = 128; off > 0; off >>= 1) {
        if (t < off) red[t] += red[t + off];
        __syncthreads();
    }
    if (t == 0) c0[0] = red[0] + b4[0];
}

// ---------------------------------------------------------------------------
// 3) f2 = wordFeatures @ W2^T : (80,300)x(300,1024). Tiny.
// ---------------------------------------------------------------------------
__global__ void k_fc2(const float* __restrict__ wf, const float* __restrict__ W2,
                      float* __restrict__ f2) {
    int idx = blockIdx.x * blockDim.x + threadIdx.x;    // KK*DD
    int k = idx / DD, d = idx % DD;
    const float* wr  = wf + (size_t)k * WDIM;
    const float* w2r = W2 + (size_t)d * WDIM;
    float acc = 0.f;
    for (int w = 0; w < WDIM; ++w) acc = fmaf(wr[w], w2r[w], acc);
    f2[idx] = acc;
}

// ---------------------------------------------------------------------------
// 4) f1 = xT @ W1^T via V_WMMA_F32_16X16X4_F32, exact f32 math.
//    256 threads = 8 waves, 64x64 C-tile, K staged in chunks of 16 through
//    DOUBLE-BUFFERED LDS filled by async global->LDS b128 copies; the next
//    chunk's fill (2 async ops/wave) overlaps the current chunk's 8 WMMAs,
//    retired with s_wait_asynccnt 2.
//    LDS layout [row][20]: A-frag pairs are contiguous (ds_load_b64) and the
//    20-float pitch + 0..3 column offsets hit all 64 banks exactly once.
//    A frag (ISA 7.12.2): lanes 0-15 K=+0/+1, lanes 16-31 K=+2/+3.
//    B frag: VGPR0 rows K=+0(lo)/+1(hi), VGPR1 rows K=+2(lo)/+3(hi).
// ---------------------------------------------------------------------------
__global__ __launch_bounds__(256) void k_fc1_wmma(const float* __restrict__ xT,
                                                  const float* __restrict__ W1,
                                                  float* __restrict__ f1) {
    __shared__ float As[2][64][20];
    __shared__ float Bs[2][64][20];
    int tid  = threadIdx.x;
    int wave = tid >> 5;
    int lane = tid & 31;
    int l    = lane & 15;
    int hiA  = (lane >> 4) << 1;   // 0 or 2
    int hiB  = (lane >> 4);        // 0 or 1
    int m0 = blockIdx.x * 64;
    int n0 = blockIdx.y * 64;
    int mi  = wave & 3;            // subtile row 0..3
    int ni0 = (wave >> 2) << 1;    // subtile col pair 0 or 2

    // Fill mapping: thread -> (row fi, 4-col group fj); one b128 per array.
    int fi = tid >> 2;             // 0..63
    int fj = (tid & 3) << 2;       // 0,4,8,12
    int mc = m0 + fi;
    if (mc > MM - 1) mc = MM - 1;  // clamp tail rows (their C rows never stored)
    const float* gA = xT + (size_t)mc * CC + fj;
    const float* gB = W1 + (size_t)(n0 + fi) * CC + fj;
    unsigned ldsA[2] = { (unsigned)(size_t)&As[0][fi][fj], (unsigned)(size_t)&As[1][fi][fj] };
    unsigned ldsB[2] = { (unsigned)(size_t)&Bs[0][fi][fj], (unsigned)(size_t)&Bs[1][fi][fj] };

    v8f acc0 = {};
    v8f acc1 = {};

    // Prologue: fill buffer 0 with chunk 0.
    async_ld128(ldsA[0], gA);
    async_ld128(ldsB[0], gB);

    const int NCHUNK = CC / 16;    // 128
    for (int c = 0; c < NCHUNK; ++c) {
        int cur = c & 1;
        if (c + 1 < NCHUNK) {
            int nxt = cur ^ 1;     // == buffer of chunk c-1, freed by last barrier
            async_ld128(ldsA[nxt], gA + (size_t)(c + 1) * 16);
            async_ld128(ldsB[nxt], gB + (size_t)(c + 1) * 16);
            wait_async_le2();      // chunk c's two ops complete; c+1 in flight
        } else {
            wait_async_le();
        }
        __syncthreads();           // all waves' fills of buffer `cur` visible
        #pragma unroll
        for (int q = 0; q < 4; ++q) {
            int ka = q * 4 + hiA;
            int kb = q * 4 + hiB;
            v2f a, b0, b1;
            a[0]  = As[cur][mi * 16 + l][ka];
            a[1]  = As[cur][mi * 16 + l][ka + 1];
            b0[0] = Bs[cur][ni0 * 16 + l][kb];
            b0[1] = Bs[cur][ni0 * 16 + l][kb + 2];
            b1[0] = Bs[cur][(ni0 + 1) * 16 + l][kb];
            b1[1] = Bs[cur][(ni0 + 1) * 16 + l][kb + 2];
            acc0 = __builtin_amdgcn_wmma_f32_16x16x4_f32(false, a, false, b0,
                                                         (short)0, acc0, false, false);
            acc1 = __builtin_amdgcn_wmma_f32_16x16x4_f32(false, a, false, b1,
                                                         (short)0, acc1, false, false);
        }
        __syncthreads();           // buffer `cur` free for refill at chunk c+2
    }

    // C/D layout: VGPR r -> lanes 0-15: M=r, lanes 16-31: M=8+r; N=lane%16.
    int mrow_base = m0 + mi * 16 + ((lane >= 16) ? 8 : 0);
    #pragma unroll
    for (int r = 0; r < 8; ++r) {
        int m = mrow_base + r;
        if (m < MM) {
            f1[(size_t)m * DD + n0 + ni0 * 16 + l]       = acc0[r];
            f1[(size_t)m * DD + n0 + (ni0 + 1) * 16 + l] = acc1[r];
        }
    }
}

// ---------------------------------------------------------------------------
// 5) Fused tanh-dot: logits[m,k] = sum_d tanh(f1[m,d]*f2[k,d])*v[d] + c0.
//    One wave per (m,k); b128 loads; native v_tanh_f32 when available.
// ---------------------------------------------------------------------------
__global__ void k_logits(const float* __restrict__ f1, const float* __restrict__ f2,
                         const float* __restrict__ v, const float* __restrict__ c0,
                         float* __restrict__ coef) {
    int gw   = blockIdx.x * (blockDim.x >> 5) + (threadIdx.x >> 5);
    int lane = threadIdx.x & 31;
    int m = gw / KK;
    int k = gw % KK;
    const v4f* f1r = (const v4f*)(f1 + (size_t)m * DD);
    const v4f* f2r = (const v4f*)(f2 + (size_t)k * DD);
    const v4f* vv  = (const v4f*)v;
    float s = 0.f;
    #pragma unroll
    for (int d = lane; d < DD / 4; d += 32) {      // 8 iterations
        v4f a = f1r[d], b = f2r[d], w = vv[d];
        s = fmaf(FAST_TANH(a[0] * b[0]), w[0], s);
        s = fmaf(FAST_TANH(a[1] * b[1]), w[1], s);
        s = fmaf(FAST_TANH(a[2] * b[2]), w[2], s);
        s = fmaf(FAST_TANH(a[3] * b[3]), w[3], s);
    }
    #pragma unroll
    for (int off = 16; off > 0; off >>= 1) s += __shfl_xor(s, off, 32);
    if (lane == 0) coef[(size_t)m * KK + k] = s + c0[0];
}

// ---------------------------------------------------------------------------
// 6) Softmax over 196 spatial positions per (b,k), in place; one wave each.
// ---------------------------------------------------------------------------
__global__ void k_softmax(float* __restrict__ coef) {
    int gw   = blockIdx.x * (blockDim.x >> 5) + (threadIdx.x >> 5);  // 0..639
    int lane = threadIdx.x & 31;
    int b = gw / KK, k = gw % KK;
    float* base = coef + (size_t)b * HW * KK + k;
    float mx = -1e30f;
    for (int hw = lane; hw < HW; hw += 32) mx = fmaxf(mx, base[(size_t)hw * KK]);
    #pragma unroll
    for (int off = 16; off > 0; off >>= 1) mx = fmaxf(mx, __shfl_xor(mx, off, 32));
    float sum = 0.f;
    for (int hw = lane; hw < HW; hw += 32) sum += __expf(base[(size_t)hw * KK] - mx);
    #pragma unroll
    for (int off = 16; off > 0; off >>= 1) sum += __shfl_xor(sum, off, 32);
    float inv = 1.f / sum;
    for (int hw = lane; hw < HW; hw += 32)
        base[(size_t)hw * KK] = __expf(base[(size_t)hw * KK] - mx) * inv;
}

// ---------------------------------------------------------------------------
// 7) Streaming finale (roofline: ~1.03 GB written):
//    fwc[b,hw,k,c] = xT[b,hw,c]*coef[b,hw,k] via non-temporal b128 stores;
//    sf[b,k,c] accumulated in registers. Each thread owns 4 consecutive c.
// ---------------------------------------------------------------------------
__global__ __launch_bounds__(256) void k_fwc_sf(const float* __restrict__ xT,
                                                const float* __restrict__ coef,
                                                float* __restrict__ fwc,
                                                float* __restrict__ sf) {
    __shared__ float cf[HW];
    int b = blockIdx.z, k = blockIdx.y;
    int c4 = (blockIdx.x * 256 + threadIdx.x) * 4;   // gridDim.x = CC/1024
    for (int hw = threadIdx.x; hw < HW; hw += 256)
        cf[hw] = coef[((size_t)b * HW + hw) * KK + k];
    __syncthreads();
    v4f acc = {};
    for (int hw = 0; hw < HW; ++hw) {
        v4f xv = *(const v4f*)&xT[((size_t)b * HW + hw) * CC + c4];
        float s = cf[hw];
        v4f val = xv * s;
        __builtin_nontemporal_store(val,
            (v4f*)&fwc[(((size_t)b * HW + hw) * KK + k) * CC + c4]);
        acc += val;
    }
    *(v4f*)&sf[((size_t)b * KK + k) * CC + c4] = acc;
}

// ---------------------------------------------------------------------------
extern "C" void kernel_launch(void* const* d_in, const int* in_sizes, int n_in,
                              void* d_out, int out_size, void* d_ws, size_t ws_size,
                              hipStream_t stream) {
    const float* img = (const float*)d_in[0];   // (B,C,14,14)
    const float* wf  = (const float*)d_in[1];   // (K,WD)
    const float* W1  = (const float*)d_in[2];   // (D,C)
    const float* W2  = (const float*)d_in[3];   // (D,WD)
    const float* W3  = (const float*)d_in[4];   // (D,D)
    const float* b3  = (const float*)d_in[5];   // (D,)
    const float* W4  = (const float*)d_in[6];   // (1,D)
    const float* b4  = (const float*)d_in[7];   // (1,)

    // Output layout: semanticFeature (B,K,C) | fwc (B,HW,K,C) | coef (B,HW,K)
    float* out  = (float*)d_out;
    float* sf   = out;
    float* fwc  = out + (size_t)BB * KK * CC;
    float* coef = fwc + (size_t)BB * HW * KK * CC;

    // Workspace layout (floats): ~19.6 MB total, all 16B-aligned offsets.
    float* ws = (float*)d_ws;
    size_t o = 0;
    float* xT = ws + o;  o += (size_t)BB * HW * CC;   // 3,211,264
    float* f1 = ws + o;  o += (size_t)MM * DD;        // 1,605,632
    float* f2 = ws + o;  o += (size_t)KK * DD;        //    81,920
    float* v  = ws + o;  o += DD;                     //     1,024
    float* c0 = ws + o;  o += 4;

    k_transpose<<<dim3((HW + 31) / 32, CC / 32, BB), dim3(32, 8), 0, stream>>>(img, xT);
    k_foldW34 <<<DD / 256, 256, 0, stream>>>(W3, W4, v);
    k_c0      <<<1, 256, 0, stream>>>(b3, W4, b4, c0);
    k_fc2     <<<(KK * DD) / 256, 256, 0, stream>>>(wf, W2, f2);
    k_fc1_wmma<<<dim3((MM + 63) / 64, DD / 64), 256, 0, stream>>>(xT, W1, f1);
    k_logits  <<<(MM * KK) / 8, 256, 0, stream>>>(f1, f2, v, c0, coef);
    k_softmax <<<(BB * KK) / 8, 256, 0, stream>>>(coef);
    k_fwc_sf  <<<dim3(CC / 1024, KK, BB), 256, 0, stream>>>(xT, coef, fwc, sf);
}